// NLLPMFLoss_75634374083218
// MI455X (gfx1250) — compile-verified
//
#include <hip/hip_runtime.h>

typedef __attribute__((ext_vector_type(2))) float v2f;
typedef __attribute__((ext_vector_type(8))) float v8f;

#define EPSF 1e-9f
#define TDIM 32
#define ROWSTRIDE_B 144                  // 128B row data + 16B pad, 16B aligned
#define TILE_ROWS 16
#define TILE_LDS_B (TILE_ROWS * ROWSTRIDE_B)   // 2304 B per wave tile buffer
#define WAVES_PER_BLOCK 8
#define BLOCK 256

// Issue 4 fully-coalesced async B128 loads (512B each) for one 16x32 f32 tile
// into the wave's LDS buffer. Always issues exactly 4 instructions so that
// ASYNCcnt arithmetic for pipelined waits is exact; out-of-range rows are
// clamped to the tile's row 0 (in-bounds) and excluded from accumulation later.
__device__ __forceinline__ void issue_tile_async(const float* __restrict__ preds,
                                                 unsigned lds_base,
                                                 long long row0, int lane, int N)
{
    const unsigned long long gbase =
        (unsigned long long)(uintptr_t)preds + (unsigned long long)row0 * (TDIM * 4ull);
#pragma unroll
    for (int c = 0; c < 4; ++c) {
        const int id = c * 32 + lane;        // 16B chunk id within 2KB tile
        const int rr = id >> 3;
        const int kk = id & 7;
        const unsigned ldso = lds_base + (unsigned)(rr * ROWSTRIDE_B + kk * 16);
        const unsigned long long goff =
            (row0 + rr < (long long)N) ? (unsigned long long)id * 16ull
                                       : (unsigned long long)kk * 16ull; // clamp into row0
        const unsigned long long ga = gbase + goff;
        asm volatile("global_load_async_to_lds_b128 %0, %1, off"
                     :: "v"(ldso), "v"(ga) : "memory");
    }
}

__global__ __launch_bounds__(BLOCK)
void nllpmf_main(const float* __restrict__ preds,
                 const int*   __restrict__ targets,
                 const float* __restrict__ weight,
                 const float* __restrict__ sweight,
                 float* __restrict__ partials,
                 int N, int ntiles)
{
    __shared__ float s_w[TDIM];
    __shared__ float s_wc[TDIM];
    __shared__ __align__(16) char s_tile[2 * WAVES_PER_BLOCK * TILE_LDS_B]; // ping-pong
    __shared__ float s_red[2 * WAVES_PER_BLOCK];

    const int tid  = threadIdx.x;
    const int lane = tid & 31;
    const int wid  = tid >> 5;

    // per-block weight table + prefix means (tiny, once)
    if (tid < TDIM) s_w[tid] = weight[tid];
    __syncthreads();
    if (tid < TDIM) {
        float s = 0.f;
        for (int k = 0; k <= tid; ++k) s += s_w[k];
        s_wc[tid] = s / (float)(tid + 1);
    }
    __syncthreads();

    char* tileA = s_tile + (2 * wid + 0) * TILE_LDS_B;
    char* tileB = s_tile + (2 * wid + 1) * TILE_LDS_B;
    const unsigned baseA = (unsigned)(uintptr_t)tileA; // low 32 bits of flat == LDS offset
    const unsigned baseB = (unsigned)(uintptr_t)tileB;

    const int r = lane & 15;   // row within tile owned by this lane
    const int h = lane >> 4;   // K-pair half, k%4 in {2h, 2h+1} (WMMA A-operand layout)

    float accn = 0.f, accd = 0.f;

    const int gw     = blockIdx.x * WAVES_PER_BLOCK + wid;
    const int stride = gridDim.x * WAVES_PER_BLOCK;

    const v2f bones = {1.f, 1.f}; // all-ones B operand (layout-immune)

    int buf = 0;
    if (gw < ntiles)
        issue_tile_async(preds, baseA, (long long)gw * TILE_ROWS, lane, N);

    for (int tile = gw; tile < ntiles; tile += stride) {
        const long long row0 = (long long)tile * TILE_ROWS;
        char* curtile = buf ? tileB : tileA;

        // ---- software pipeline: prefetch next tile, then wait for current ----
        const int nexttile = tile + stride;
        if (nexttile < ntiles) {
            issue_tile_async(preds, buf ? baseA : baseB,
                             (long long)nexttile * TILE_ROWS, lane, N);
            asm volatile("s_wait_asynccnt 4" ::: "memory"); // oldest 4 (current tile) done
        } else {
            asm volatile("s_wait_asynccnt 0" ::: "memory");
        }

        // ---- read my row slice in WMMA-A interleave: K = 4c+2h, 4c+2h+1 ----
        const float* lrow = (const float*)(curtile + r * ROWSTRIDE_B);
        float ex[16];
#pragma unroll
        for (int c = 0; c < 8; ++c) {
            float2 v = *(const float2*)(lrow + 4 * c + 2 * h); // 8B aligned, conflict-free
            ex[2 * c]     = v.x;
            ex[2 * c + 1] = v.y;
        }

        // ---- row max (lane half + partner lane) ----
        float m = ex[0];
#pragma unroll
        for (int i = 1; i < 16; ++i) m = fmaxf(m, ex[i]);
        m = fmaxf(m, __shfl_xor(m, 16, 32));

#pragma unroll
        for (int i = 0; i < 16; ++i) ex[i] = __expf(ex[i] - m);

        // ---- Z = row sums of exp via f32 WMMA: D = EX(16x32) * ones(32x16) ----
        v8f zacc = {0.f, 0.f, 0.f, 0.f, 0.f, 0.f, 0.f, 0.f};
#pragma unroll
        for (int c = 0; c < 8; ++c) {
            v2f a = {ex[2 * c], ex[2 * c + 1]};
            zacc = __builtin_amdgcn_wmma_f32_16x16x4_f32(
                false, a, false, bones, (short)0, zacc, false, false);
        }
        // zacc[q] = Z_{q+8h} broadcast across the lane half; extract Z_r
        float z0  = (r & 1) ? zacc[1] : zacc[0];
        float z1  = (r & 1) ? zacc[3] : zacc[2];
        float z2  = (r & 1) ? zacc[5] : zacc[4];
        float z3  = (r & 1) ? zacc[7] : zacc[6];
        float z01 = (r & 2) ? z1 : z0;
        float z23 = (r & 2) ? z3 : z2;
        float zsel = (r & 4) ? z23 : z01;
        float zoth = __shfl_xor(zsel, 16, 32);
        float Z = (((r >> 3) & 1) == h) ? zsel : zoth;
        float invZ = 1.0f / Z;

        // ---- per-row target / sample weight ----
        const long long row = row0 + r;
        const bool valid = row < (long long)N;
        int d = 0, e = 0; float sw = 0.f;
        if (valid) {
            d  = targets[2 * row];
            e  = targets[2 * row + 1];
            sw = sweight[row];
        }
        d = min(max(d, 0), TDIM - 1);

        // ---- cdf[d] = masked sum of clipped pmf (lane half + partner) ----
        float cs = 0.f;
#pragma unroll
        for (int c = 0; c < 8; ++c) {
            const int k0 = 4 * c + 2 * h;
            float p0 = fminf(fmaxf(ex[2 * c]     * invZ, EPSF), 1.f - EPSF);
            float p1 = fminf(fmaxf(ex[2 * c + 1] * invZ, EPSF), 1.f - EPSF);
            if (k0     <= d) cs += p0;
            if (k0 + 1 <= d) cs += p1;
        }
        const float cdf_d = cs + __shfl_xor(cs, 16, 32);

        // ---- pmf[d] via one LDS re-read; loss terms ----
        const float xd = lrow[d];
        const float pd = fminf(fmaxf(__expf(xd - m) * invZ, EPSF), 1.f - EPSF);
        const float Sd = fminf(fmaxf(1.f - cdf_d, EPSF), 1.f);
        const float ell = e ? (-__logf(pd) * s_w[d]) : (-__logf(Sd) * s_wc[d]);

        if (valid && h == 0) {          // one contribution per row
            accn += ell * sw;
            accd += sw;
        }

        buf ^= 1;
    }

    // ---- deterministic block reduction ----
#pragma unroll
    for (int off = 16; off > 0; off >>= 1) {
        accn += __shfl_xor(accn, off, 32);
        accd += __shfl_xor(accd, off, 32);
    }
    if (lane == 0) { s_red[2 * wid] = accn; s_red[2 * wid + 1] = accd; }
    __syncthreads();
    if (tid == 0) {
        float n0 = 0.f, d0 = 0.f;
        for (int w = 0; w < WAVES_PER_BLOCK; ++w) { n0 += s_red[2 * w]; d0 += s_red[2 * w + 1]; }
        partials[2 * blockIdx.x]     = n0;
        partials[2 * blockIdx.x + 1] = d0;
    }
}

__global__ __launch_bounds__(256)
void nllpmf_finalize(const float* __restrict__ partials, int count, float* __restrict__ out)
{
    __shared__ float sn[256], sd[256];
    float a = 0.f, b = 0.f;
    for (int i = threadIdx.x; i < count; i += 256) {
        a += partials[2 * i];
        b += partials[2 * i + 1];
    }
    sn[threadIdx.x] = a; sd[threadIdx.x] = b;
    __syncthreads();
    for (int s = 128; s > 0; s >>= 1) {
        if ((int)threadIdx.x < s) {
            sn[threadIdx.x] += sn[threadIdx.x + s];
            sd[threadIdx.x] += sd[threadIdx.x + s];
        }
        __syncthreads();
    }
    if (threadIdx.x == 0) out[0] = sn[0] / fmaxf(sd[0], EPSF);
}

extern "C" void kernel_launch(void* const* d_in, const int* in_sizes, int n_in,
                              void* d_out, int out_size, void* d_ws, size_t ws_size,
                              hipStream_t stream)
{
    const float* preds   = (const float*)d_in[0];
    const int*   targets = (const int*)  d_in[1];
    const float* weight  = (const float*)d_in[2];
    const float* sweight = (const float*)d_in[3];

    const int N = in_sizes[3];                 // sample_weight has N elements
    const int ntiles = (N + TILE_ROWS - 1) / TILE_ROWS;

    int grid = 2048;
    const size_t need = (size_t)grid * 2 * sizeof(float);
    if (ws_size < need) grid = (int)(ws_size / (2 * sizeof(float)));
    if (grid < 1) grid = 1;
    if (grid > ntiles) grid = (ntiles + WAVES_PER_BLOCK - 1) / WAVES_PER_BLOCK;
    if (grid < 1) grid = 1;

    float* partials = (float*)d_ws;

    nllpmf_main<<<grid, BLOCK, 0, stream>>>(preds, targets, weight, sweight,
                                            partials, N, ntiles);
    nllpmf_finalize<<<1, 256, 0, stream>>>(partials, grid, (float*)d_out);
}